// GlobalNegativeContrastiveLoss_5274219840045
// MI455X (gfx1250) — compile-verified
//
#include <hip/hip_runtime.h>

#define B_   512
#define D_   512
#define C_   1000
#define M_   200
#define KP   50
#define KN   20
#define NCOL (C_ * KN)        // 20000 negative columns
#define NBLK 313              // ceil(NCOL / 64)
#define NSLOT (NBLK * 2)      // per-row deterministic partial slots
#define INV_T (1.0f / 0.07f)
#define MARG 0.5f

typedef float v2f __attribute__((ext_vector_type(2)));
typedef float v8f __attribute__((ext_vector_type(8)));

// D = A(16x4, f32) * B(4x16, f32) + C(16x16, f32)  -- fp32 tensor op on CDNA5
__device__ __forceinline__ v8f wmma4(v2f a, v2f b, v8f c) {
  return __builtin_amdgcn_wmma_f32_16x16x4_f32(
      /*neg_a=*/false, a, /*neg_b=*/false, b,
      /*c_mod=*/(short)0, c, /*reuse_a=*/false, /*reuse_b=*/false);
}

// ---------------------------------------------------------------- init ov map
__global__ void k_init(int* __restrict__ ov) {
  int i = blockIdx.x * blockDim.x + threadIdx.x;
  if (i < C_ * KP) ov[i] = -1;
}

// ------------------------------------------------------------- L2 normalize f
__global__ void k_norm(const float* __restrict__ feat, float* __restrict__ f) {
  int b = blockIdx.x, t = threadIdx.x;           // 256 threads
  __shared__ float red[256];
  __shared__ float inv;
  float x0 = feat[b * D_ + t];
  float x1 = feat[b * D_ + t + 256];
  red[t] = x0 * x0 + x1 * x1;
  __syncthreads();
  for (int st = 128; st > 0; st >>= 1) {
    if (t < st) red[t] += red[t + st];
    __syncthreads();
  }
  if (t == 0) inv = 1.0f / fmaxf(sqrtf(red[0]), 1e-12f);
  __syncthreads();
  f[b * D_ + t]       = x0 * inv;
  f[b * D_ + t + 256] = x1 * inv;
}

// ------------------------------- rank / group size / first index / ov scatter
__global__ void k_rank(const int* __restrict__ labels, const int* __restrict__ mptr,
                       int* __restrict__ rank, int* __restrict__ gsz,
                       int* __restrict__ fio, int* __restrict__ ov) {
  int b = threadIdx.x;                           // one block of 512 threads
  int lab = labels[b];
  int r = 0, g = 0, fi = B_;
  for (int j = 0; j < B_; ++j) {
    if (labels[j] == lab) {
      ++g;
      if (j < b) ++r;
      if (j < fi) fi = j;
    }
  }
  rank[b] = r; gsz[b] = g; fio[b] = fi;
  int wp = (mptr[lab] + r) % M_;
  if (wp < KP) ov[lab * KP + wp] = b;            // unique slot per sample
}

// ------------------------------------------------------------- G = f * f^T
__global__ void k_gram(const float* __restrict__ f, float* __restrict__ G) {
  int gw = (blockIdx.x * blockDim.x + threadIdx.x) >> 5;  // global wave id
  int l  = threadIdx.x & 31;
  int tm = (gw >> 5) << 4;                       // 32x32 tiles of 16x16
  int tn = (gw & 31) << 4;
  int mr = l & 15;
  int ko = (l >> 4) * 2;                         // lane-half K offset
  const float* arow = f + (size_t)(tm + mr) * D_;
  const float* brow = f + (size_t)(tn + mr) * D_;
  v8f acc = {};
  for (int k = 0; k < D_; k += 4) {
    v2f a = *(const v2f*)(arow + k + ko);
    v2f b = *(const v2f*)(brow + k + ko);
    acc = wmma4(a, b, acc);
  }
#pragma unroll
  for (int r = 0; r < 8; ++r) {
    int row = tm + r + ((l >> 4) << 3);
    G[(size_t)row * B_ + tn + mr] = acc[r];
  }
}

// --------------- negatives GEMM: f (512xD) x bank[:, :20]^T (D x 20000), fused
__global__ void k_neg(const float* __restrict__ f, const float* __restrict__ bank,
                      const int* __restrict__ labels, const int* __restrict__ ov,
                      const float* __restrict__ G,
                      float* __restrict__ allPart, float* __restrict__ hardPart,
                      int* __restrict__ cntPart) {
  __shared__ float As[64 * 33];                  // 64 rows x 32 k, stride 33
  __shared__ float Bs[64 * 33];                  // 64 cols x 32 k, stride 33
  const int t = threadIdx.x;                     // 256 threads = 8 waves
  const int w = t >> 5, l = t & 31;
  const int mwave = (w & 3) * 16;                // 4 waves along M
  const int nwave = (w >> 2) * 32;               // 2 waves along N (2x16 tiles)
  const int rowBase = blockIdx.y * 64;
  const int colBase = blockIdx.x * 64;
  const int mr = l & 15;
  const int ko = (l >> 4) * 2;

  v8f acc0 = {}, acc1 = {};

  for (int k0 = 0; k0 < D_; k0 += 32) {
#pragma unroll
    for (int i = 0; i < 8; ++i) {                // stage A tile
      int idx = t + i * 256;
      int r = idx >> 5, k = idx & 31;
      As[r * 33 + k] = f[(size_t)(rowBase + r) * D_ + k0 + k];
    }
#pragma unroll
    for (int i = 0; i < 8; ++i) {                // stage B tile (bank rows)
      int idx = t + i * 256;
      int cc = idx >> 5, k = idx & 31;
      int j = colBase + cc;
      float v = 0.0f;
      if (j < NCOL) {
        int c = j / KN, kk = j % KN;
        const float* p = bank + ((size_t)c * M_ + kk) * D_ + k0 + k;
        v = *p;
        if (k0 + 32 < D_) __builtin_prefetch(p + 32, 0, 0);
      }
      Bs[cc * 33 + k] = v;
    }
    __syncthreads();
#pragma unroll
    for (int kk = 0; kk < 32; kk += 4) {
      v2f a, b0, b1;
      a.x  = As[(mwave + mr) * 33 + kk + ko];
      a.y  = As[(mwave + mr) * 33 + kk + ko + 1];
      b0.x = Bs[(nwave + mr) * 33 + kk + ko];
      b0.y = Bs[(nwave + mr) * 33 + kk + ko + 1];
      b1.x = Bs[(nwave + 16 + mr) * 33 + kk + ko];
      b1.y = Bs[(nwave + 16 + mr) * 33 + kk + ko + 1];
      acc0 = wmma4(a, b0, acc0);
      acc1 = wmma4(a, b1, acc1);
    }
    __syncthreads();
  }

  // fused epilogue: override patch, class mask, margin, per-row reduction
#pragma unroll
  for (int r = 0; r < 8; ++r) {
    int row = rowBase + mwave + r + ((l >> 4) << 3);
    int lab = labels[row];
    float aS = 0.0f, hS = 0.0f;
    int hC = 0;
#pragma unroll
    for (int half = 0; half < 2; ++half) {
      float v = (half == 0) ? acc0[r] : acc1[r];
      int col = colBase + nwave + half * 16 + mr;
      if (col < NCOL) {
        int c = col / KN, k = col % KN;
        int o = ov[c * KP + k];
        if (o >= 0) v = G[(size_t)row * B_ + o];  // slot was overwritten by f[o]
        float s = v * INV_T;
        if (c != lab) {
          aS += s;
          if (s > MARG) { hS += s; ++hC; }
        }
      }
    }
    // reduce across the 16 lanes that share this row
    for (int m = 1; m < 16; m <<= 1) {
      aS += __shfl_xor(aS, m, 16);
      hS += __shfl_xor(hS, m, 16);
      hC += __shfl_xor(hC, m, 16);
    }
    if (mr == 0) {
      size_t slot = (size_t)row * NSLOT + (size_t)blockIdx.x * 2 + (w >> 2);
      allPart[slot]  = aS;
      hardPart[slot] = hS;
      cntPart[slot]  = hC;
    }
  }
}

// ---------------------------------------------- positives: batch part + own-class
__global__ void k_pos(const float* __restrict__ f, const float* __restrict__ G,
                      const float* __restrict__ bank, const int* __restrict__ labels,
                      const int* __restrict__ rank, const int* __restrict__ gsz,
                      const int* __restrict__ fio, const int* __restrict__ ov,
                      float* __restrict__ posLoss) {
  int b = blockIdx.x, t = threadIdx.x;           // 256 threads
  __shared__ float sred[256];
  __shared__ int cred[256];
  __shared__ int ovk[KP];
  int lab = labels[b];
  int fi = fio[b];
  int g = gsz[b];
  if (t < KP) ovk[t] = ov[lab * KP + t];
  __syncthreads();

  float s = 0.0f;
  int cnt = 0;
  // batch positives from Gram matrix
  for (int j = t; j < B_; j += 256) {
    if (labels[j] == lab && rank[j] != fi && g > 1) {
      s += G[(size_t)b * B_ + j];
      ++cnt;
    }
  }
  // own-class memory positives (non-overridden rows): dot(f_b, sum_k row_k)
  for (int d = t; d < D_; d += 256) {
    float fd = f[(size_t)b * D_ + d];
    float rs = 0.0f;
    for (int k = 0; k < KP; ++k)
      if (ovk[k] < 0) rs += bank[((size_t)lab * M_ + k) * D_ + d];
    s += fd * rs;
  }
  sred[t] = s; cred[t] = cnt;
  __syncthreads();
  for (int st = 128; st > 0; st >>= 1) {
    if (t < st) { sred[t] += sred[t + st]; cred[t] += cred[t + st]; }
    __syncthreads();
  }
  if (t == 0) {
    float tot = sred[0];
    for (int k = 0; k < KP; ++k)
      if (ovk[k] >= 0) tot += G[(size_t)b * B_ + ovk[k]];  // overridden slots
    float pos_sum = tot * INV_T;
    float pos_cnt = (float)(cred[0] + KP);
    posLoss[b] = -pos_sum / pos_cnt;
  }
}

// ------------------------------------ per-row partial reduction -> loss per b
__global__ void k_rowred(const float* __restrict__ allPart, const float* __restrict__ hardPart,
                         const int* __restrict__ cntPart, const float* __restrict__ posLoss,
                         float* __restrict__ lossB) {
  int b = blockIdx.x, t = threadIdx.x;           // 256 threads
  __shared__ float sa[256], sh[256];
  __shared__ int sc[256];
  float a = 0.0f, h = 0.0f;
  int c = 0;
  for (int i = t; i < NSLOT; i += 256) {
    size_t s = (size_t)b * NSLOT + i;
    a += allPart[s]; h += hardPart[s]; c += cntPart[s];
  }
  sa[t] = a; sh[t] = h; sc[t] = c;
  __syncthreads();
  for (int st = 128; st > 0; st >>= 1) {
    if (t < st) { sa[t] += sa[t + st]; sh[t] += sh[t + st]; sc[t] += sc[t + st]; }
    __syncthreads();
  }
  if (t == 0) {
    float nl = (sc[0] > 0) ? (sh[0] / (float)sc[0])
                           : (sa[0] / (float)((C_ - 1) * KN));
    lossB[b] = posLoss[b] + nl;
  }
}

// ------------------------------------------------------------------ final mean
__global__ void k_final(const float* __restrict__ lossB, float* __restrict__ out) {
  __shared__ float red[B_];
  int t = threadIdx.x;                           // 512 threads
  red[t] = lossB[t];
  __syncthreads();
  for (int st = 256; st > 0; st >>= 1) {
    if (t < st) red[t] += red[t + st];
    __syncthreads();
  }
  if (t == 0) out[0] = red[0] / (float)B_;
}

extern "C" void kernel_launch(void* const* d_in, const int* in_sizes, int n_in,
                              void* d_out, int out_size, void* d_ws, size_t ws_size,
                              hipStream_t stream) {
  (void)in_sizes; (void)n_in; (void)out_size; (void)ws_size;
  const float* features = (const float*)d_in[0];
  const int*   labels   = (const int*)d_in[1];
  const float* bank     = (const float*)d_in[2];
  const int*   mptr     = (const int*)d_in[3];

  char* w = (char*)d_ws;
  float* f        = (float*)w; w += (size_t)B_ * D_ * sizeof(float);
  float* G        = (float*)w; w += (size_t)B_ * B_ * sizeof(float);
  int*   rank     = (int*)w;   w += (size_t)B_ * sizeof(int);
  int*   gsz      = (int*)w;   w += (size_t)B_ * sizeof(int);
  int*   fio      = (int*)w;   w += (size_t)B_ * sizeof(int);
  int*   ov       = (int*)w;   w += (size_t)C_ * KP * sizeof(int);
  float* allPart  = (float*)w; w += (size_t)B_ * NSLOT * sizeof(float);
  float* hardPart = (float*)w; w += (size_t)B_ * NSLOT * sizeof(float);
  int*   cntPart  = (int*)w;   w += (size_t)B_ * NSLOT * sizeof(int);
  float* posLoss  = (float*)w; w += (size_t)B_ * sizeof(float);
  float* lossB    = (float*)w; w += (size_t)B_ * sizeof(float);

  k_init<<<(C_ * KP + 255) / 256, 256, 0, stream>>>(ov);
  k_norm<<<B_, 256, 0, stream>>>(features, f);
  k_rank<<<1, B_, 0, stream>>>(labels, mptr, rank, gsz, fio, ov);
  k_gram<<<128, 256, 0, stream>>>(f, G);
  k_neg<<<dim3(NBLK, 8), 256, 0, stream>>>(f, bank, labels, ov, G,
                                           allPart, hardPart, cntPart);
  k_pos<<<B_, 256, 0, stream>>>(f, G, bank, labels, rank, gsz, fio, ov, posLoss);
  k_rowred<<<B_, 256, 0, stream>>>(allPart, hardPart, cntPart, posLoss, lossB);
  k_final<<<1, B_, 0, stream>>>(lossB, (float*)d_out);
}